// SuperGAT_82738249990425
// MI455X (gfx1250) — compile-verified
//
#include <hip/hip_runtime.h>
#include <math.h>

// SuperGAT (2-layer, MX attention) for MI455X / gfx1250, wave32.
// GEMMs use V_WMMA_F32_16X16X4_F32 (exact fp32). Edge attention is
// wave-per-edge with L2-resident gathers (node features = 51MB < 192MB L2).

typedef float v2f __attribute__((ext_vector_type(2)));
typedef float v8f __attribute__((ext_vector_type(8)));

#define NEG_SLOPE 0.2f

// ---- monotone int key for float atomicMax ----
__device__ __forceinline__ int fkey(float f) {
  int i = __float_as_int(f);
  return i >= 0 ? i : (i ^ 0x7FFFFFFF);
}
__device__ __forceinline__ float kinv(int k) {
  return __int_as_float(k >= 0 ? k : (k ^ 0x7FFFFFFF));
}
__device__ __forceinline__ float sigmoidf_(float x) {
  return 1.0f / (1.0f + __expf(-x));
}
__device__ __forceinline__ float lrelu(float x) {
  return x > 0.0f ? x : NEG_SLOPE * x;
}

// ============================= init kernels =============================
__global__ void init_l1(float* __restrict__ agg, float* __restrict__ den,
                        int* __restrict__ mx, int n) {
  int tid = blockIdx.x * blockDim.x + threadIdx.x;
  if (tid < n * 128) agg[tid] = 0.0f;
  if (tid < n * 8) { den[tid] = 0.0f; mx[tid] = (int)0x80000000; }
}

__global__ void init_l2(float* __restrict__ out, float* __restrict__ den,
                        int* __restrict__ mx, int n) {
  int tid = blockIdx.x * blockDim.x + threadIdx.x;
  if (tid < n * 40) out[tid] = 0.0f;
  if (tid < n) { den[tid] = 0.0f; mx[tid] = (int)0x80000000; }
}

// ========================= GEMM1: H1 = X @ W1 ===========================
// X:[N,128] W:[128,128] H:[N,128].  256 thr = 8 waves; wave w -> ntile w.
__global__ __launch_bounds__(256) void gemm1_wmma(
    const float* __restrict__ X, const float* __restrict__ W,
    float* __restrict__ H) {
  __shared__ float xs[16 * 128];
  const int mtile = blockIdx.x;
  const int wave = threadIdx.x >> 5;
  const int lane = threadIdx.x & 31;
  const long row0 = (long)mtile * 16;

  // stage 16x128 A tile in LDS (512 float4 / 256 threads)
  {
    const float4* src = reinterpret_cast<const float4*>(X + row0 * 128);
    float4* dst = reinterpret_cast<float4*>(xs);
    dst[threadIdx.x] = src[threadIdx.x];
    dst[threadIdx.x + 256] = src[threadIdx.x + 256];
  }
  __syncthreads();

  const int m = lane & 15;
  const int kb = (lane >> 4) * 2;          // 0 or 2 (half-wave K offset)
  const int n = wave * 16 + (lane & 15);   // global output column
  v8f acc = {};
#pragma unroll 4
  for (int k0 = 0; k0 < 128; k0 += 4) {
    v2f a, b;
    a.x = xs[m * 128 + k0 + kb];
    a.y = xs[m * 128 + k0 + kb + 1];
    b.x = W[(k0 + kb) * 128 + n];
    b.y = W[(k0 + kb + 1) * 128 + n];
    acc = __builtin_amdgcn_wmma_f32_16x16x4_f32(false, a, false, b,
                                                (short)0, acc, false, false);
  }
  const int half = lane >> 4;
#pragma unroll
  for (int v = 0; v < 8; ++v) {
    int mr = v + 8 * half;
    H[(row0 + mr) * 128 + n] = acc[v];
  }
}

// ==================== GEMM2: G2 = H2in @ W2 (40 cols) ===================
// 128 thr = 4 waves; waves 0..2 compute ntiles covering cols 0..47 (pad).
__global__ __launch_bounds__(128) void gemm2_wmma(
    const float* __restrict__ X, const float* __restrict__ W,
    float* __restrict__ G) {
  __shared__ float xs[16 * 128];
  const int mtile = blockIdx.x;
  const int wave = threadIdx.x >> 5;
  const int lane = threadIdx.x & 31;
  const long row0 = (long)mtile * 16;
  {
    const float4* src = reinterpret_cast<const float4*>(X + row0 * 128);
    float4* dst = reinterpret_cast<float4*>(xs);
#pragma unroll
    for (int i = 0; i < 4; ++i) dst[threadIdx.x + i * 128] = src[threadIdx.x + i * 128];
  }
  __syncthreads();
  if (wave >= 3) return;                   // wave-uniform: EXEC all-1 in WMMA waves

  const int m = lane & 15;
  const int kb = (lane >> 4) * 2;
  const int n = wave * 16 + (lane & 15);   // 0..47, valid cols < 40
  v8f acc = {};
#pragma unroll 4
  for (int k0 = 0; k0 < 128; k0 += 4) {
    v2f a, b;
    a.x = xs[m * 128 + k0 + kb];
    a.y = xs[m * 128 + k0 + kb + 1];
    b.x = (n < 40) ? W[(k0 + kb) * 40 + n] : 0.0f;
    b.y = (n < 40) ? W[(k0 + kb + 1) * 40 + n] : 0.0f;
    acc = __builtin_amdgcn_wmma_f32_16x16x4_f32(false, a, false, b,
                                                (short)0, acc, false, false);
  }
  if (n < 40) {
    const int half = lane >> 4;
#pragma unroll
    for (int v = 0; v < 8; ++v) {
      int mr = v + 8 * half;
      G[(row0 + mr) * 40 + n] = acc[v];
    }
  }
}

// ================= per-node attention projections =======================
__global__ void attvec1(const float* __restrict__ H,
                        const float* __restrict__ attL, const float* __restrict__ attR,
                        float* __restrict__ aL, float* __restrict__ aR, int n) {
  int tid = blockIdx.x * blockDim.x + threadIdx.x;
  if (tid >= n * 8) return;
  int node = tid >> 3, h = tid & 7;
  const float* hp = H + (long)node * 128 + h * 16;
  const float* lp = attL + h * 16;
  const float* rp = attR + h * 16;
  float sl = 0.0f, sr = 0.0f;
#pragma unroll
  for (int c = 0; c < 16; ++c) { float v = hp[c]; sl += v * lp[c]; sr += v * rp[c]; }
  aL[tid] = sl;
  aR[tid] = sr;
}

__global__ void attvec2(const float* __restrict__ G,
                        const float* __restrict__ attL, const float* __restrict__ attR,
                        float* __restrict__ aL, float* __restrict__ aR, int n) {
  int tid = blockIdx.x * blockDim.x + threadIdx.x;
  if (tid >= n) return;
  const float* gp = G + (long)tid * 40;
  float sl = 0.0f, sr = 0.0f;
#pragma unroll
  for (int c = 0; c < 40; ++c) { float v = gp[c]; sl += v * attL[c]; sr += v * attR[c]; }
  aL[tid] = sl;
  aR[tid] = sr;
}

// ================= layer-1 edge pass A: alpha + segment max =============
// wave per edge; lane handles 4 channels; head = lane>>2 (4 lanes/head).
__global__ __launch_bounds__(256) void edge1a(
    const int* __restrict__ ei, const float* __restrict__ H,
    const float* __restrict__ aL, const float* __restrict__ aR,
    float* __restrict__ alpha, int* __restrict__ mx, int E, int n) {
  const int wave = threadIdx.x >> 5, lane = threadIdx.x & 31;
  const long e = (long)blockIdx.x * 8 + wave;
  const long E2 = (long)E + n;
  if (e >= E2) return;
  int src, dst;
  if (e < E) { src = ei[e]; dst = ei[(long)E + e]; } else { src = dst = (int)(e - E); }

  float4 hj = *reinterpret_cast<const float4*>(H + (long)src * 128 + lane * 4);
  float4 hi = *reinterpret_cast<const float4*>(H + (long)dst * 128 + lane * 4);
  float p = hj.x * hi.x + hj.y * hi.y + hj.z * hi.z + hj.w * hi.w;
  p += __shfl_xor(p, 1, 32);
  p += __shfl_xor(p, 2, 32);               // 4-lane head group reduced
  const int h = lane >> 2;
  float a = (aL[(long)src * 8 + h] + aR[(long)dst * 8 + h]) * sigmoidf_(p);
  a = lrelu(a);
  if ((lane & 3) == 0) {
    alpha[e * 8 + h] = a;
    atomicMax(&mx[(long)dst * 8 + h], fkey(a));
  }
}

// === layer-1 edge pass B: exp, denom, unnormalized aggregate (atomic) ===
__global__ __launch_bounds__(256) void edge1b(
    const int* __restrict__ ei, const float* __restrict__ H,
    const float* __restrict__ alpha, const int* __restrict__ mx,
    float* __restrict__ den, float* __restrict__ agg, int E, int n) {
  const int wave = threadIdx.x >> 5, lane = threadIdx.x & 31;
  const long e = (long)blockIdx.x * 8 + wave;
  const long E2 = (long)E + n;
  if (e >= E2) return;
  int src, dst;
  if (e < E) { src = ei[e]; dst = ei[(long)E + e]; } else { src = dst = (int)(e - E); }

  const int h = lane >> 2;
  float ex = __expf(alpha[e * 8 + h] - kinv(mx[(long)dst * 8 + h]));
  if ((lane & 3) == 0) atomicAdd(&den[(long)dst * 8 + h], ex);
  float4 hj = *reinterpret_cast<const float4*>(H + (long)src * 128 + lane * 4);
  float* ap = agg + (long)dst * 128 + lane * 4;
  atomicAdd(ap + 0, hj.x * ex);
  atomicAdd(ap + 1, hj.y * ex);
  atomicAdd(ap + 2, hj.z * ex);
  atomicAdd(ap + 3, hj.w * ex);
}

// ======= layer-1 epilogue: normalize, +bias, ELU (in place on agg) ======
__global__ void node1(float* __restrict__ agg, const float* __restrict__ den,
                      const float* __restrict__ b1, int n) {
  int tid = blockIdx.x * blockDim.x + threadIdx.x;
  if (tid >= n * 128) return;
  int node = tid >> 7, c = tid & 127, h = c >> 4;
  float v = agg[tid] / den[(long)node * 8 + h] + b1[c];
  agg[tid] = v > 0.0f ? v : expm1f(v);     // ELU
}

// ================= layer-2 edge pass A (1 head, 40 ch) ==================
__global__ __launch_bounds__(256) void edge2a(
    const int* __restrict__ ei, const float* __restrict__ G,
    const float* __restrict__ aL, const float* __restrict__ aR,
    float* __restrict__ alpha, int* __restrict__ mx, int E, int n) {
  const int wave = threadIdx.x >> 5, lane = threadIdx.x & 31;
  const long e = (long)blockIdx.x * 8 + wave;
  const long E2 = (long)E + n;
  if (e >= E2) return;
  int src, dst;
  if (e < E) { src = ei[e]; dst = ei[(long)E + e]; } else { src = dst = (int)(e - E); }

  const float* gj = G + (long)src * 40;
  const float* gi = G + (long)dst * 40;
  float p = gj[lane] * gi[lane];
  if (lane < 8) p += gj[32 + lane] * gi[32 + lane];
  p += __shfl_xor(p, 16, 32);
  p += __shfl_xor(p, 8, 32);
  p += __shfl_xor(p, 4, 32);
  p += __shfl_xor(p, 2, 32);
  p += __shfl_xor(p, 1, 32);
  if (lane == 0) {
    float a = lrelu((aL[src] + aR[dst]) * sigmoidf_(p));
    alpha[e] = a;
    atomicMax(&mx[dst], fkey(a));
  }
}

// ================= layer-2 edge pass B (accumulate into d_out) ==========
__global__ __launch_bounds__(256) void edge2b(
    const int* __restrict__ ei, const float* __restrict__ G,
    const float* __restrict__ alpha, const int* __restrict__ mx,
    float* __restrict__ den, float* __restrict__ out, int E, int n) {
  const int wave = threadIdx.x >> 5, lane = threadIdx.x & 31;
  const long e = (long)blockIdx.x * 8 + wave;
  const long E2 = (long)E + n;
  if (e >= E2) return;
  int src, dst;
  if (e < E) { src = ei[e]; dst = ei[(long)E + e]; } else { src = dst = (int)(e - E); }

  float ex = __expf(alpha[e] - kinv(mx[dst]));
  if (lane == 0) atomicAdd(&den[dst], ex);
  const float* gj = G + (long)src * 40;
  float* op = out + (long)dst * 40;
  atomicAdd(&op[lane], gj[lane] * ex);
  if (lane < 8) atomicAdd(&op[32 + lane], gj[32 + lane] * ex);
}

// ======= final: normalize, +b2, log_softmax (in place, wave/node) =======
__global__ __launch_bounds__(256) void final_lsm(
    float* __restrict__ out, const float* __restrict__ den,
    const float* __restrict__ b2, int n) {
  const int wave = threadIdx.x >> 5, lane = threadIdx.x & 31;
  const long node = (long)blockIdx.x * 8 + wave;
  if (node >= n) return;
  float inv = 1.0f / den[node];
  float* op = out + node * 40;
  float v0 = op[lane] * inv + b2[lane];
  float v1 = (lane < 8) ? (op[32 + lane] * inv + b2[32 + lane])
                        : __int_as_float(0xFF800000);  // -inf
  float m = fmaxf(v0, v1);
  m = fmaxf(m, __shfl_xor(m, 16, 32));
  m = fmaxf(m, __shfl_xor(m, 8, 32));
  m = fmaxf(m, __shfl_xor(m, 4, 32));
  m = fmaxf(m, __shfl_xor(m, 2, 32));
  m = fmaxf(m, __shfl_xor(m, 1, 32));
  float s = __expf(v0 - m) + ((lane < 8) ? __expf(v1 - m) : 0.0f);
  s += __shfl_xor(s, 16, 32);
  s += __shfl_xor(s, 8, 32);
  s += __shfl_xor(s, 4, 32);
  s += __shfl_xor(s, 2, 32);
  s += __shfl_xor(s, 1, 32);
  float lse = m + __logf(s);
  op[lane] = v0 - lse;
  if (lane < 8) op[32 + lane] = v1 - lse;
}

// ================================ launch ================================
extern "C" void kernel_launch(void* const* d_in, const int* in_sizes, int n_in,
                              void* d_out, int out_size, void* d_ws, size_t ws_size,
                              hipStream_t stream) {
  const float* x     = (const float*)d_in[0];
  const int*   ei    = (const int*)  d_in[1];
  const float* W1    = (const float*)d_in[2];
  const float* attL1 = (const float*)d_in[3];
  const float* attR1 = (const float*)d_in[4];
  const float* b1    = (const float*)d_in[5];
  const float* W2    = (const float*)d_in[6];
  const float* attL2 = (const float*)d_in[7];
  const float* attR2 = (const float*)d_in[8];
  const float* b2    = (const float*)d_in[9];
  float* out = (float*)d_out;

  const int N = in_sizes[0] / 128;
  const int E = in_sizes[1] / 2;
  const long E2 = (long)E + N;

  float* ws = (float*)d_ws;
  // layer-1 layout
  float* H1    = ws;                                  // N*128
  float* AGG1  = H1 + (size_t)N * 128;                // N*128 (becomes H2in)
  float* ALPH1 = AGG1 + (size_t)N * 128;              // E2*8
  float* aL1   = ALPH1 + (size_t)E2 * 8;              // N*8
  float* aR1   = aL1 + (size_t)N * 8;                 // N*8
  int*   MAX1  = (int*)(aR1 + (size_t)N * 8);         // N*8
  float* DEN1  = (float*)(MAX1 + (size_t)N * 8);      // N*8
  // layer-2 layout overlays H1 region (H1 dead after edge1b)
  float* G2    = H1;                                  // N*40
  float* aL2   = G2 + (size_t)N * 40;                 // N
  float* aR2   = aL2 + N;                             // N
  int*   MAX2  = (int*)(aR2 + N);                     // N
  float* DEN2  = (float*)(MAX2 + N);                  // N
  float* ALPH2 = ALPH1;                               // E2

  const int eb = (int)((E2 + 7) / 8);                 // 8 waves (edges) / block
  const int mtiles = (N + 15) / 16;                   // 6250 (N divisible by 16)

  // ---- layer 1 ----
  init_l1<<<(N * 128 + 255) / 256, 256, 0, stream>>>(AGG1, DEN1, MAX1, N);
  gemm1_wmma<<<mtiles, 256, 0, stream>>>(x, W1, H1);
  attvec1<<<(N * 8 + 255) / 256, 256, 0, stream>>>(H1, attL1, attR1, aL1, aR1, N);
  edge1a<<<eb, 256, 0, stream>>>(ei, H1, aL1, aR1, ALPH1, MAX1, E, N);
  edge1b<<<eb, 256, 0, stream>>>(ei, H1, ALPH1, MAX1, DEN1, AGG1, E, N);
  node1<<<(N * 128 + 255) / 256, 256, 0, stream>>>(AGG1, DEN1, b1, N);

  // ---- layer 2 ----
  init_l2<<<(N * 40 + 255) / 256, 256, 0, stream>>>(out, DEN2, MAX2, N);
  gemm2_wmma<<<mtiles, 128, 0, stream>>>(AGG1, W2, G2);
  attvec2<<<(N + 255) / 256, 256, 0, stream>>>(G2, attL2, attR2, aL2, aR2, N);
  edge2a<<<eb, 256, 0, stream>>>(ei, G2, aL2, aR2, ALPH2, MAX2, E, N);
  edge2b<<<eb, 256, 0, stream>>>(ei, G2, ALPH2, MAX2, DEN2, out, E, N);
  final_lsm<<<(N + 7) / 8, 256, 0, stream>>>(out, DEN2, b2, N);
}